// GraphPPOAgent_87857851007455
// MI455X (gfx1250) — compile-verified
//
#include <hip/hip_runtime.h>

// ---------------------------------------------------------------------------
// CDNA5 (gfx1250) heterogeneous GINE forward.  fp32 WMMA (V_WMMA_F32_16X16X4_F32)
// for all dense 64-wide GEMMs; weights pre-swizzled to fragment-major layout so
// every B fragment is a single b64 load; fused edge-embed + scatter with f32
// atomics.  Full tiles use unguarded stores with immediate-offset addressing;
// only the ragged firewall tail (25000 % 16 == 8) runs the guarded path.
// ---------------------------------------------------------------------------

typedef float v2f __attribute__((ext_vector_type(2)));
typedef float v8f __attribute__((ext_vector_type(8)));

#define WMMA_F32X4(a, b, c) \
  __builtin_amdgcn_wmma_f32_16x16x4_f32(false, (a), false, (b), (short)0, (c), false, false)

#define HID 64

static __device__ __forceinline__ float relu_f(float v) {
  return __builtin_fmaxf(v, 0.0f);
}
static __device__ __forceinline__ v2f ld2(const float* p) {
  return *(const v2f*)p;
}

// ----------------------------- utility kernels -----------------------------

__global__ void zero_kernel(float* __restrict__ p, size_t n) {
  size_t i = (size_t)blockIdx.x * blockDim.x + threadIdx.x;
  size_t stride = (size_t)gridDim.x * blockDim.x;
  for (; i < n; i += stride) p[i] = 0.0f;
}

__global__ void relu_kernel(float* __restrict__ p, size_t n) {
  size_t i = (size_t)blockIdx.x * blockDim.x + threadIdx.x;
  size_t stride = (size_t)gridDim.x * blockDim.x;
  for (; i < n; i += stride) p[i] = relu_f(p[i]);
}

// b'[n] = b[n] + sum_{k} W[(koff+k)*64 + n]   (folds the concat-ones half)
__global__ void bias_fold_kernel(const float* __restrict__ W,
                                 const float* __restrict__ b,
                                 float* __restrict__ out, int krows, int koff) {
  int n = threadIdx.x;  // 64 threads
  float s = b[n];
  for (int k = 0; k < krows; ++k) s += W[(size_t)(koff + k) * HID + n];
  out[n] = s;
}

// xd["switch"] = relu(broadcast(switch_emb))
__global__ void switch_init_kernel(const float* __restrict__ emb,
                                   float* __restrict__ xs, size_t total) {
  size_t i = (size_t)blockIdx.x * blockDim.x + threadIdx.x;
  size_t stride = (size_t)gridDim.x * blockDim.x;
  for (; i < total; i += stride) xs[i] = relu_f(emb[i & 63]);
}

// ------------------- weight repack: row-major -> fragment-major -------------
// Wp[((nt*nkk + kk)*32 + lane)*2 + {0,1}] = W[(kk*4 + koff(lane)) + {0,1}][nt*16 + (lane&15)]
// so each lane's B fragment for WMMA step (kk, nt) is one contiguous float2.
__global__ void repack_w_kernel(const float* __restrict__ W,
                                float* __restrict__ Wp, int nkk) {
  int total = 128 * nkk;  // 4 nt * nkk * 32 lanes
  int tid = blockIdx.x * blockDim.x + threadIdx.x;
  if (tid >= total) return;
  int nt = tid / (nkk * 32);
  int rem = tid % (nkk * 32);
  int kk = rem / 32;
  int lane = rem % 32;
  int m = lane & 15;
  int koff = (lane >> 4) * 2;
  Wp[(size_t)tid * 2 + 0] = W[(size_t)(kk * 4 + koff) * HID + nt * 16 + m];
  Wp[(size_t)tid * 2 + 1] = W[(size_t)(kk * 4 + koff + 1) * HID + nt * 16 + m];
}

// --------------------------- node linear (WMMA) ----------------------------
// out[row] = relu( x[row, 0:16] @ W + bias64 ); one wave / 16 rows; Wp packed.
template <bool FULL>
__global__ __launch_bounds__(32) void node_lin_kernel(
    const float* __restrict__ x, const float* __restrict__ Wp,
    const float* __restrict__ bias, float* __restrict__ out, int N,
    int row_base) {
  int lane = threadIdx.x;
  int m = lane & 15;
  int half = lane >> 4;  // 0: K lanes {0,1}; 1: K lanes {2,3}
  int r0 = row_base + blockIdx.x * 16;
  int row = r0 + m;
  if (!FULL && row >= N) row = N - 1;  // clamp reads; writes guarded in tail

  const v2f* xr2 = (const v2f*)(x + (size_t)row * 16);
  const v2f* wp2 = (const v2f*)Wp;
  v2f a[4];
#pragma unroll
  for (int kk = 0; kk < 4; ++kk) a[kk] = xr2[kk * 2 + half];
  int n = m;
  // single per-lane base; all 32 outputs at compile-time-constant offsets
  float* ob = out + (size_t)(r0 + half * 8) * HID + n;
#pragma unroll
  for (int nt = 0; nt < 4; ++nt) {
    float bv = bias[nt * 16 + n];
    v8f c = {bv, bv, bv, bv, bv, bv, bv, bv};
#pragma unroll
    for (int kk = 0; kk < 4; ++kk)
      c = WMMA_F32X4(a[kk], wp2[(nt * 4 + kk) * 32 + lane], c);
#pragma unroll
    for (int j = 0; j < 8; ++j) {
      if (FULL) {
        ob[nt * 16 + (size_t)j * HID] = relu_f(c[j]);
      } else {
        int rr = r0 + j + half * 8;
        if (rr < N) out[(size_t)rr * HID + nt * 16 + n] = relu_f(c[j]);
      }
    }
  }
}

// ----------------- fused edge-embed + gather + relu + scatter ---------------
// agg[dst[e]] += relu( xsrc[src[e]] + (ea[e] @ We[0:8,:] + eb) )
// one thread per (edge, channel-pair); a wave covers all 64 channels of one
// edge, so ea reads are wave-broadcast and We/x/agg accesses are coalesced.
__global__ void msg_scatter_kernel(const float* __restrict__ xsrc,
                                   const float* __restrict__ ea,
                                   const int* __restrict__ src,
                                   const int* __restrict__ dst,
                                   const float* __restrict__ We,
                                   const float* __restrict__ eb,
                                   float* __restrict__ agg, int E) {
  size_t tid = (size_t)blockIdx.x * blockDim.x + threadIdx.x;
  int e = (int)(tid >> 5);
  int cp = (int)(tid & 31) * 2;  // channel pair
  if (e >= E) return;
  int s = src[e];
  int d = dst[e];
  v2f v = ld2(eb + cp);
#pragma unroll
  for (int k = 0; k < 8; ++k) {
    float t = ea[(size_t)e * 8 + k];
    v2f w = ld2(We + (size_t)k * HID + cp);
    v.x += t * w.x;
    v.y += t * w.y;
  }
  v2f xv = ld2(xsrc + (size_t)s * HID + cp);
  float m0 = relu_f(v.x + xv.x);
  float m1 = relu_f(v.y + xv.y);
  float* ap = agg + (size_t)d * HID + cp;
  unsafeAtomicAdd(ap + 0, m0);
  unsafeAtomicAdd(ap + 1, m1);
}

// -------------------------- GINE MLP (WMMA, fused) --------------------------
// out[row] += relu( (x[row]+agg[row]) @ W1 + b1 ) @ W2 + b2
// One wave per 16-row tile: GEMM1 (K=64, 16 WMMA steps x 4 N-tiles), relu
// staged through LDS (stride 68 -> bank-conflict-free b64 reads), GEMM2 same.
#define HSTR 68
template <bool FULL>
__global__ __launch_bounds__(32) void mlp_acc_kernel(
    const float* __restrict__ x, const float* __restrict__ agg,
    const float* __restrict__ W1p, const float* __restrict__ b1,
    const float* __restrict__ W2p, const float* __restrict__ b2,
    float* __restrict__ out, int N, int row_base) {
  __shared__ float hid[16 * HSTR];
  int lane = threadIdx.x;
  int m = lane & 15;
  int half = lane >> 4;
  int r0 = row_base + blockIdx.x * 16;
  int row = r0 + m;
  if (!FULL && row >= N) row = N - 1;
  const v2f* xr2 = (const v2f*)(x + (size_t)row * HID);
  const v2f* ar2 = (const v2f*)(agg + (size_t)row * HID);
  const v2f* w1 = (const v2f*)W1p;
  const v2f* w2 = (const v2f*)W2p;

  // A fragments for GEMM1: h = x + agg (b64 loads, immediate offsets)
  v2f a1[16];
#pragma unroll
  for (int kk = 0; kk < 16; ++kk) {
    v2f xa = xr2[kk * 2 + half];
    v2f aa = ar2[kk * 2 + half];
    a1[kk].x = xa.x + aa.x;
    a1[kk].y = xa.y + aa.y;
  }
  int n = m;
#pragma unroll
  for (int nt = 0; nt < 4; ++nt) {
    float bv = b1[nt * 16 + n];
    v8f c = {bv, bv, bv, bv, bv, bv, bv, bv};
#pragma unroll
    for (int kk = 0; kk < 16; ++kk)
      c = WMMA_F32X4(a1[kk], w1[(nt * 16 + kk) * 32 + lane], c);
    // relu + stage hidden tile to LDS in row-major (rows j / j+8 per C layout)
#pragma unroll
    for (int j = 0; j < 8; ++j)
      hid[(j + half * 8) * HSTR + nt * 16 + n] = relu_f(c[j]);
  }
  __syncthreads();

  // GEMM2: hidden @ W2 + b2, accumulate into out
  float* ob = out + (size_t)(r0 + half * 8) * HID + n;  // constant-offset base
#pragma unroll
  for (int nt = 0; nt < 4; ++nt) {
    float bv = b2[nt * 16 + n];
    v8f c = {bv, bv, bv, bv, bv, bv, bv, bv};
#pragma unroll
    for (int kk = 0; kk < 16; ++kk) {
      v2f aa = *(const v2f*)&hid[m * HSTR + kk * 4 + half * 2];  // ds_load_b64
      c = WMMA_F32X4(aa, w2[(nt * 16 + kk) * 32 + lane], c);
    }
#pragma unroll
    for (int j = 0; j < 8; ++j) {
      if (FULL) {
        ob[nt * 16 + (size_t)j * HID] += c[j];
      } else {
        int rr = r0 + j + half * 8;
        if (rr < N) out[(size_t)rr * HID + nt * 16 + n] += c[j];
      }
    }
  }
}

// ------------------------------- host driver -------------------------------

extern "C" void kernel_launch(void* const* d_in, const int* in_sizes, int n_in,
                              void* d_out, int out_size, void* d_ws, size_t ws_size,
                              hipStream_t stream) {
  (void)out_size; (void)ws_size;
  // node type order used for buffer layout == d_out concat order:
  // router(50k), host(100k), firewall(25k), switch(100k)
  enum { T_ROUTER = 0, T_HOST = 1, T_FIREWALL = 2, T_SWITCH = 3 };
  static const long rowOff[4] = {0, 50000, 150000, 175000};
  static const int cnt[4] = {50000, 100000, 25000, 100000};
  const long NTOT = 275000;

  // ---- workspace carve-up (floats) ----
  float* Xf = (float*)d_ws;            // 275000*64 : layer input features
  float* OUT1 = Xf + NTOT * HID;       // 275000*64 : conv1 output / conv2 input
  float* AGG = OUT1 + NTOT * HID;      // 100000*64 : per-relation aggregate
  float* NB = AGG + 100000L * HID;     // 3*64 : folded node biases (r,h,f)
  float* EB = NB + 3 * HID;            // 64   : folded edge bias
  float* WPC = EB + HID;               // 24*4096 : packed conv weights
  float* WPN = WPC + 24 * 4096;        // 3*1024 : packed node weights

  // ---- inputs ----
  const float* x_node[3] = {(const float*)d_in[0], (const float*)d_in[1],
                            (const float*)d_in[2]};  // router, host, firewall

  // relations in reference order: sr, rs, sh, hs, sf, fs.
  // perm = index of relation name under sorted-key pytree flatten
  // (sorted: fs,hs,rs,sf,sh,sr).
  struct RelMeta { int ea, si, di, st, dt, perm; };
  const RelMeta rels[6] = {
      {3, 4, 5, T_SWITCH, T_ROUTER, 5},       // sr
      {6, 7, 8, T_ROUTER, T_SWITCH, 2},       // rs
      {9, 10, 11, T_SWITCH, T_HOST, 4},       // sh
      {12, 13, 14, T_HOST, T_SWITCH, 1},      // hs
      {15, 16, 17, T_SWITCH, T_FIREWALL, 3},  // sf
      {18, 19, 20, T_FIREWALL, T_SWITCH, 0},  // fs
  };

  // ---- params: 57 pytree leaves, sorted-key flatten order:
  // conv1{fs,hs,rs,sf,sh,sr}(W1,b1,W2,b2), conv2{...}, edge_lin(W,b),
  // node_lin{firewall,host,router}(W,b), switch_emb
  const int NLEAF = 57;
  const float* leaf[NLEAF];
  int leafsz[NLEAF];
  {
    int li = 0;
    for (int c = 0; c < 2; ++c)
      for (int r = 0; r < 6; ++r) {
        leafsz[li++] = 4096; leafsz[li++] = 64;
        leafsz[li++] = 4096; leafsz[li++] = 64;
      }
    leafsz[li++] = 1024; leafsz[li++] = 64;  // edge_lin
    for (int t = 0; t < 3; ++t) { leafsz[li++] = 2048; leafsz[li++] = 64; }
    leafsz[li++] = 64;  // switch_emb
  }
  if (n_in >= 21 + NLEAF) {
    for (int i = 0; i < NLEAF; ++i) leaf[i] = (const float*)d_in[21 + i];
  } else {
    // params passed as one concatenated blob
    const float* base = (const float*)d_in[21];
    long off = 0;
    for (int i = 0; i < NLEAF; ++i) { leaf[i] = base + off; off += leafsz[i]; }
  }
  const float* We = leaf[48];
  const float* be = leaf[49];
  // node_lin sorted: firewall(50,51), host(52,53), router(54,55)
  const float* Wn[3] = {leaf[54], leaf[52], leaf[50]};  // router, host, firewall
  const float* bn[3] = {leaf[55], leaf[53], leaf[51]};
  const float* semb = leaf[56];

  const size_t NF = (size_t)NTOT * HID;
  const int ZB = 2048;  // grid-stride blocks for big memset/relu

  // ---- fold the concat-ones halves into biases ----
  for (int t = 0; t < 3; ++t)
    hipLaunchKernelGGL(bias_fold_kernel, dim3(1), dim3(64), 0, stream,
                       Wn[t], bn[t], NB + t * HID, 16, 16);
  hipLaunchKernelGGL(bias_fold_kernel, dim3(1), dim3(64), 0, stream,
                     We, be, EB, 8, 8);

  // ---- pre-swizzle weights into fragment-major layout ----
  for (int t = 0; t < 3; ++t)  // node weights (K=16 -> nkk=4, 512 frag-pairs)
    hipLaunchKernelGGL(repack_w_kernel, dim3(2), dim3(256), 0, stream,
                       Wn[t], WPN + t * 1024, 4);
  for (int layer = 0; layer < 2; ++layer)
    for (int r = 0; r < 6; ++r) {
      int pbase = layer * 24 + 4 * rels[r].perm;
      float* slot = WPC + (size_t)(layer * 12 + r * 2) * 4096;
      hipLaunchKernelGGL(repack_w_kernel, dim3(8), dim3(256), 0, stream,
                         leaf[pbase + 0], slot, 16);          // W1
      hipLaunchKernelGGL(repack_w_kernel, dim3(8), dim3(256), 0, stream,
                         leaf[pbase + 2], slot + 4096, 16);   // W2
    }

  // ---- node encoders -> Xf ----
  for (int t = 0; t < 3; ++t) {
    int full = cnt[t] / 16, rem = cnt[t] % 16;
    if (full)
      hipLaunchKernelGGL((node_lin_kernel<true>), dim3(full), dim3(32), 0,
                         stream, x_node[t], WPN + t * 1024, NB + t * HID,
                         Xf + rowOff[t] * HID, cnt[t], 0);
    if (rem)
      hipLaunchKernelGGL((node_lin_kernel<false>), dim3(1), dim3(32), 0,
                         stream, x_node[t], WPN + t * 1024, NB + t * HID,
                         Xf + rowOff[t] * HID, cnt[t], full * 16);
  }
  hipLaunchKernelGGL(switch_init_kernel, dim3(ZB), dim3(256), 0, stream, semb,
                     Xf + rowOff[T_SWITCH] * HID, (size_t)cnt[T_SWITCH] * HID);

  // ---- two GINE conv layers ----
  for (int layer = 0; layer < 2; ++layer) {
    const float* xin = (layer == 0) ? Xf : OUT1;
    float* xout = (layer == 0) ? OUT1 : (float*)d_out;
    hipLaunchKernelGGL(zero_kernel, dim3(ZB), dim3(256), 0, stream, xout, NF);
    for (int r = 0; r < 6; ++r) {
      const RelMeta& R = rels[r];
      int E = in_sizes[R.si];  // 500000
      int Nd = cnt[R.dt];
      hipLaunchKernelGGL(zero_kernel, dim3(ZB), dim3(256), 0, stream, AGG,
                         (size_t)Nd * HID);
      size_t threads = (size_t)E * 32;
      hipLaunchKernelGGL(msg_scatter_kernel,
                         dim3((unsigned)((threads + 255) / 256)), dim3(256), 0,
                         stream, xin + rowOff[R.st] * HID,
                         (const float*)d_in[R.ea], (const int*)d_in[R.si],
                         (const int*)d_in[R.di], We, EB, AGG, E);
      int pbase = layer * 24 + 4 * R.perm;
      const float* W1p = WPC + (size_t)(layer * 12 + r * 2) * 4096;
      const float* W2p = W1p + 4096;
      int full = Nd / 16, rem = Nd % 16;
      if (full)
        hipLaunchKernelGGL((mlp_acc_kernel<true>), dim3(full), dim3(32), 0,
                           stream, xin + rowOff[R.dt] * HID, AGG, W1p,
                           leaf[pbase + 1], W2p, leaf[pbase + 3],
                           xout + rowOff[R.dt] * HID, Nd, 0);
      if (rem)
        hipLaunchKernelGGL((mlp_acc_kernel<false>), dim3(1), dim3(32), 0,
                           stream, xin + rowOff[R.dt] * HID, AGG, W1p,
                           leaf[pbase + 1], W2p, leaf[pbase + 3],
                           xout + rowOff[R.dt] * HID, Nd, full * 16);
    }
    hipLaunchKernelGGL(relu_kernel, dim3(ZB), dim3(256), 0, stream, xout, NF);
  }
}